// Subjective_GCN_88441966559606
// MI455X (gfx1250) — compile-verified
//
#include <hip/hip_runtime.h>
#include <math.h>

typedef __attribute__((ext_vector_type(2))) float v2f;
typedef __attribute__((ext_vector_type(8))) float v8f;

#define F_IN 256
#define CH   16
#define NL   7

// ---------------- degree / normalization ----------------

__global__ void k_degree(const int* __restrict__ dst, unsigned* __restrict__ deg, int E) {
    int e = blockIdx.x * blockDim.x + threadIdx.x;
    if (e < E) atomicAdd(&deg[dst[e]], 1u);
}

__global__ void k_dinv(const unsigned* __restrict__ deg, float* __restrict__ dinv, int N) {
    int i = blockIdx.x * blockDim.x + threadIdx.x;
    if (i < N) dinv[i] = rsqrtf((float)(deg[i] + 1u));   // +1 = self-loop; deg>=1 always
}

// ---------------- GEMM1: h1[N,16] = x[N,256] @ W0[256,16] via f32 WMMA ----------------
// One wave computes a 16-node x 16-channel tile, K-loop of 64 x (16x16x4) WMMAs.

__global__ void k_gemm1(const float* __restrict__ x, const float* __restrict__ W0,
                        float* __restrict__ h1, int N, int tiles) {
    int wave = (int)((blockIdx.x * blockDim.x + threadIdx.x) >> 5);
    int lane = threadIdx.x & 31;
    if (wave >= tiles) return;                 // wave-uniform exit (EXEC stays all-1 for WMMA)
    int lo = lane & 15;
    int hi = lane >> 4;
    int nb = wave * 16;

    int row = nb + lo; if (row >= N) row = N - 1;       // clamp for tail tile loads
    const float* xr = x + (size_t)row * F_IN;

    v8f acc = {};
    for (int k0 = 0; k0 < F_IN; k0 += 4) {
        // A 16x4 f32: VGPR0 = K {0|2}, VGPR1 = K {1|3} per half-wave
        float2 av = *(const float2*)(xr + k0 + 2 * hi);
        v2f a; a.x = av.x; a.y = av.y;
        // B 4x16 f32: row K striped across lanes; half-waves split K-range
        v2f b;
        b.x = W0[(k0 + 2 * hi)     * CH + lo];
        b.y = W0[(k0 + 2 * hi + 1) * CH + lo];
        acc = __builtin_amdgcn_wmma_f32_16x16x4_f32(false, a, false, b,
                                                    (short)0, acc, false, false);
    }

    // D: M = v + 8*hi, N = lo.  Full-tile fast path is wave-uniform (no exec churn).
    if (nb + 16 <= N) {
        float* hp = h1 + (size_t)(nb + 8 * hi) * CH + lo;
#pragma unroll
        for (int v = 0; v < 8; ++v) hp[(size_t)v * CH] = acc[v];
    } else {
#pragma unroll
        for (int v = 0; v < 8; ++v) {
            int r = nb + v + 8 * hi;
            if (r < N) h1[(size_t)r * CH + lo] = acc[v];
        }
    }
}

// ---------------- edge scatter, layer 1: agg1[dst] += h1[src] * w ----------------

__global__ void k_scatter1(const int* __restrict__ src, const int* __restrict__ dst,
                           const float* __restrict__ dinv, const float* __restrict__ h1,
                           float* __restrict__ agg1, int E) {
    int e = blockIdx.x * blockDim.x + threadIdx.x;
    if (e >= E) return;
    int s = src[e], d = dst[e];
    float w = dinv[s] * dinv[d];
    const float4* hs = (const float4*)(h1 + (size_t)s * CH);
    float* ad = agg1 + (size_t)d * CH;
#pragma unroll
    for (int j = 0; j < 4; ++j) {
        float4 v = hs[j];
        __hip_atomic_fetch_add(ad + 4 * j + 0, v.x * w, __ATOMIC_RELAXED, __HIP_MEMORY_SCOPE_AGENT);
        __hip_atomic_fetch_add(ad + 4 * j + 1, v.y * w, __ATOMIC_RELAXED, __HIP_MEMORY_SCOPE_AGENT);
        __hip_atomic_fetch_add(ad + 4 * j + 2, v.z * w, __ATOMIC_RELAXED, __HIP_MEMORY_SCOPE_AGENT);
        __hip_atomic_fetch_add(ad + 4 * j + 3, v.w * w, __ATOMIC_RELAXED, __HIP_MEMORY_SCOPE_AGENT);
    }
}

// self-loop term + ReLU (in place, elementwise)
__global__ void k_relu_selfloop(const float* __restrict__ h1, const float* __restrict__ dinv,
                                float* __restrict__ agg1, int N) {
    int idx = blockIdx.x * blockDim.x + threadIdx.x;
    if (idx >= N * CH) return;
    int i = idx >> 4;
    float di = dinv[i];
    float v = agg1[idx] + h1[idx] * di * di;
    agg1[idx] = fmaxf(v, 0.0f);
}

// ---------------- GEMM2: h2[N,8] = agg1[N,16] @ W1[16,7] (N padded 7->16) ----------------

__global__ void k_gemm2(const float* __restrict__ g, const float* __restrict__ W1,
                        float* __restrict__ h2, int N, int tiles) {
    int wave = (int)((blockIdx.x * blockDim.x + threadIdx.x) >> 5);
    int lane = threadIdx.x & 31;
    if (wave >= tiles) return;
    int lo = lane & 15;
    int hi = lane >> 4;
    int nb = wave * 16;

    int row = nb + lo; if (row >= N) row = N - 1;
    const float* gr = g + (size_t)row * CH;

    v8f acc = {};
#pragma unroll
    for (int t = 0; t < 4; ++t) {
        int k0 = 4 * t;
        float2 av = *(const float2*)(gr + k0 + 2 * hi);
        v2f a; a.x = av.x; a.y = av.y;
        v2f b;                                   // pad output channels 7..15 with zeros
        b.x = (lo < NL) ? W1[(k0 + 2 * hi)     * NL + lo] : 0.0f;
        b.y = (lo < NL) ? W1[(k0 + 2 * hi + 1) * NL + lo] : 0.0f;
        acc = __builtin_amdgcn_wmma_f32_16x16x4_f32(false, a, false, b,
                                                    (short)0, acc, false, false);
    }

    // stride-8 padded rows; column 7 is an exact 0 (B padded), columns 8..15 dropped
    if (nb + 16 <= N) {
        if (lo < 8) {
            float* hp = h2 + (size_t)(nb + 8 * hi) * 8 + lo;
#pragma unroll
            for (int v = 0; v < 8; ++v) hp[(size_t)v * 8] = acc[v];
        }
    } else {
        if (lo < 8) {
#pragma unroll
            for (int v = 0; v < 8; ++v) {
                int r = nb + v + 8 * hi;
                if (r < N) h2[(size_t)r * 8 + lo] = acc[v];
            }
        }
    }
}

// ---------------- edge scatter, layer 2: out[dst] += h2[src] * w ----------------

__global__ void k_scatter2(const int* __restrict__ src, const int* __restrict__ dst,
                           const float* __restrict__ dinv, const float* __restrict__ h2,
                           float* __restrict__ out, int E) {
    int e = blockIdx.x * blockDim.x + threadIdx.x;
    if (e >= E) return;
    int s = src[e], d = dst[e];
    float w = dinv[s] * dinv[d];
    const float4* hs = (const float4*)(h2 + (size_t)s * 8);
    float4 v0 = hs[0];
    float4 v1 = hs[1];
    float* od = out + (size_t)d * NL;
    __hip_atomic_fetch_add(od + 0, v0.x * w, __ATOMIC_RELAXED, __HIP_MEMORY_SCOPE_AGENT);
    __hip_atomic_fetch_add(od + 1, v0.y * w, __ATOMIC_RELAXED, __HIP_MEMORY_SCOPE_AGENT);
    __hip_atomic_fetch_add(od + 2, v0.z * w, __ATOMIC_RELAXED, __HIP_MEMORY_SCOPE_AGENT);
    __hip_atomic_fetch_add(od + 3, v0.w * w, __ATOMIC_RELAXED, __HIP_MEMORY_SCOPE_AGENT);
    __hip_atomic_fetch_add(od + 4, v1.x * w, __ATOMIC_RELAXED, __HIP_MEMORY_SCOPE_AGENT);
    __hip_atomic_fetch_add(od + 5, v1.y * w, __ATOMIC_RELAXED, __HIP_MEMORY_SCOPE_AGENT);
    __hip_atomic_fetch_add(od + 6, v1.z * w, __ATOMIC_RELAXED, __HIP_MEMORY_SCOPE_AGENT);
}

// self-loop term + exp + 1 (in place)
__global__ void k_final(const float* __restrict__ h2, const float* __restrict__ dinv,
                        float* __restrict__ out, int N) {
    int idx = blockIdx.x * blockDim.x + threadIdx.x;
    if (idx >= N * NL) return;
    int i = idx / NL;
    int c = idx - i * NL;
    float di = dinv[i];
    float v = out[idx] + h2[(size_t)i * 8 + c] * di * di;
    out[idx] = expf(v) + 1.0f;
}

// ---------------- launcher ----------------

extern "C" void kernel_launch(void* const* d_in, const int* in_sizes, int n_in,
                              void* d_out, int out_size, void* d_ws, size_t ws_size,
                              hipStream_t stream) {
    const float* x  = (const float*)d_in[0];
    const float* W0 = (const float*)d_in[1];
    const float* W1 = (const float*)d_in[2];
    const int*   ei = (const int*)  d_in[3];

    int N = in_sizes[0] / F_IN;
    int E = in_sizes[3] / 2;
    const int* src = ei;
    const int* dst = ei + E;
    float* out = (float*)d_out;

    // workspace layout (floats): deg[N] | dinv[N] | h1[16N] | agg1[16N] | h2[8N]  (~16.8 MB)
    float*    wsf  = (float*)d_ws;
    unsigned* deg  = (unsigned*)wsf;
    float*    dinv = wsf + (size_t)N;
    float*    h1   = wsf + 2 * (size_t)N;
    float*    agg1 = h1 + 16 * (size_t)N;
    float*    h2   = agg1 + 16 * (size_t)N;

    hipMemsetAsync(deg,  0, (size_t)N * sizeof(unsigned),      stream);
    hipMemsetAsync(agg1, 0, (size_t)N * CH * sizeof(float),    stream);
    hipMemsetAsync(out,  0, (size_t)out_size * sizeof(float),  stream);

    int tiles  = (N + 15) / 16;
    int gblk   = (tiles + 7) / 8;          // 8 waves (256 threads) per block
    int eblk   = (E + 255) / 256;

    k_degree       <<<eblk, 256, 0, stream>>>(dst, deg, E);
    k_dinv         <<<(N + 255) / 256, 256, 0, stream>>>(deg, dinv, N);
    k_gemm1        <<<gblk, 256, 0, stream>>>(x, W0, h1, N, tiles);
    k_scatter1     <<<eblk, 256, 0, stream>>>(src, dst, dinv, h1, agg1, E);
    k_relu_selfloop<<<(N * CH + 255) / 256, 256, 0, stream>>>(h1, dinv, agg1, N);
    k_gemm2        <<<gblk, 256, 0, stream>>>(agg1, W1, h2, N, tiles);
    k_scatter2     <<<eblk, 256, 0, stream>>>(src, dst, dinv, h2, out, E);
    k_final        <<<(N * NL + 255) / 256, 256, 0, stream>>>(h2, dinv, out, N);
}